// NonLinearMessagePassingLayer_62749472195028
// MI455X (gfx1250) — compile-verified
//
#include <hip/hip_runtime.h>

// ---------------------------------------------------------------------------
// NonLinearMessagePassingLayer for MI455X (gfx1250), fp32 WMMA path.
//   messages = MLP3([nodes[s] | nodes[r] | edges])     (E=640000, D=128)
//   agg      = segment_sum(messages, receivers)        (atomic f32 add, L2-resident)
//   out      = LayerNorm(nodes @ w_node + MLP3([nodes | agg]))
// Each wave owns 32 rows (two 16-row M tiles) so every B fragment feeds two
// v_wmma_f32_16x16x4_f32 ops -> 1 vmem load per WMMA instead of 2.
// ---------------------------------------------------------------------------

typedef __attribute__((ext_vector_type(2))) float v2f;
typedef __attribute__((ext_vector_type(4))) float v4f;
typedef __attribute__((ext_vector_type(8))) float v8f;

#define D_FEAT 128
#define N_NODES 10000
#define N_EDGES 640000
#define TILE_STRIDE 132          // 128 + 4 pad -> conflict-free A reads, 16B-aligned rows
#define ROWS_PER_WAVE 32         // two 16-row WMMA M tiles per wave
#define WAVES_PER_BLOCK 2
#define BLOCK_THREADS (WAVES_PER_BLOCK * 32)
#define SMEM_BYTES (WAVES_PER_BLOCK * ROWS_PER_WAVE * TILE_STRIDE * 4)   // 33792 B

__device__ __forceinline__ v8f wmma4(v2f a, v2f b, v8f c) {
  return __builtin_amdgcn_wmma_f32_16x16x4_f32(false, a, false, b, (short)0, c, false, false);
}

// 32(M) x 128(N) x 128(K) GEMM: A from padded LDS (32 x 128), B (K x 128,
// row-major) from global.  Each B fragment is reused by both M tiles.
__device__ __forceinline__ void gemm128x2(const float* Xl, const float* __restrict__ W,
                                          v8f acc0[8], v8f acc1[8], int m, int kg) {
#pragma unroll 2
  for (int kb = 0; kb < 128; kb += 4) {
    const int k0 = kb + 2 * kg;               // this half-wave's K pair
    v2f a0, a1;
    { float2 t = *(const float2*)(Xl + m * TILE_STRIDE + k0);        a0.x = t.x; a0.y = t.y; }
    { float2 t = *(const float2*)(Xl + (16 + m) * TILE_STRIDE + k0); a1.x = t.x; a1.y = t.y; }
#pragma unroll
    for (int nt = 0; nt < 8; ++nt) {
      const int n = nt * 16 + m;              // column index (lane&15)
      v2f b;
      b.x = W[k0 * 128 + n];
      b.y = W[(k0 + 1) * 128 + n];
      acc0[nt] = wmma4(a0, b, acc0[nt]);
      acc1[nt] = wmma4(a1, b, acc1[nt]);
    }
  }
}

__device__ __forceinline__ void zero_acc2(v8f acc0[8], v8f acc1[8]) {
  const v8f z = {0.f, 0.f, 0.f, 0.f, 0.f, 0.f, 0.f, 0.f};
#pragma unroll
  for (int nt = 0; nt < 8; ++nt) { acc0[nt] = z; acc1[nt] = z; }
}

// acc layout: accS[nt][v] = row (S*16 + v + 8*kg), col (nt*16 + m).
__device__ __forceinline__ void bias_relu_to_lds(float* Xl, const v8f acc0[8], const v8f acc1[8],
                                                 const float* __restrict__ bias, int m, int kg) {
#pragma unroll
  for (int nt = 0; nt < 8; ++nt) {
    const int n = nt * 16 + m;
    const float bb = bias[n];
#pragma unroll
    for (int v = 0; v < 8; ++v) {
      Xl[(v + 8 * kg) * TILE_STRIDE + n]      = fmaxf(acc0[nt][v] + bb, 0.0f);
      Xl[(16 + v + 8 * kg) * TILE_STRIDE + n] = fmaxf(acc1[nt][v] + bb, 0.0f);
    }
  }
}

// ---------------------------------------------------------------------------
__global__ void __launch_bounds__(256)
zero_ws_kernel(v4f* __restrict__ p, int n4) {
  const int i = blockIdx.x * blockDim.x + threadIdx.x;
  if (i < n4) p[i] = (v4f){0.f, 0.f, 0.f, 0.f};
}

// ---------------------------------------------------------------------------
__global__ void __launch_bounds__(BLOCK_THREADS)
edge_msg_kernel(const float* __restrict__ nodes, const float* __restrict__ edges,
                const int* __restrict__ senders, const int* __restrict__ receivers,
                const float* __restrict__ w0, const float* __restrict__ b0,
                const float* __restrict__ w1, const float* __restrict__ b1,
                const float* __restrict__ w2, const float* __restrict__ b2,
                float* __restrict__ agg) {
  extern __shared__ float smem[];
  const int lane = threadIdx.x & 31;
  const int wave = threadIdx.x >> 5;
  float* Xl = smem + wave * (ROWS_PER_WAVE * TILE_STRIDE);

  const int tile = blockIdx.x * WAVES_PER_BLOCK + wave;   // 32-edge tile
  if (tile * ROWS_PER_WAVE >= N_EDGES) return;            // wave-uniform
  const int e0 = tile * ROWS_PER_WAVE;
  const int m = lane & 15;
  const int kg = lane >> 4;

  v8f acc0[8], acc1[8];
  zero_acc2(acc0, acc1);

  // ---- layer 0, part 0: gather nodes[senders] (K = 0..127 of msg_w0) ----
  for (int r = 0; r < ROWS_PER_WAVE; ++r) {
    const v4f* src = (const v4f*)(nodes + (size_t)senders[e0 + r] * D_FEAT);
    ((v4f*)(Xl + r * TILE_STRIDE))[lane] = src[lane];
  }
  gemm128x2(Xl, w0, acc0, acc1, m, kg);

  // ---- layer 0, part 1: gather nodes[receivers] (K = 128..255) ----
  for (int r = 0; r < ROWS_PER_WAVE; ++r) {
    const v4f* src = (const v4f*)(nodes + (size_t)receivers[e0 + r] * D_FEAT);
    ((v4f*)(Xl + r * TILE_STRIDE))[lane] = src[lane];
  }
  gemm128x2(Xl, w0 + 128 * 128, acc0, acc1, m, kg);

  // ---- layer 0, part 2: edge features, streamed once -> non-temporal ----
  for (int r = 0; r < ROWS_PER_WAVE; ++r) {
    const v4f* src = (const v4f*)(edges + (size_t)(e0 + r) * D_FEAT);
    ((v4f*)(Xl + r * TILE_STRIDE))[lane] = __builtin_nontemporal_load(&src[lane]);
  }
  gemm128x2(Xl, w0 + 2 * 128 * 128, acc0, acc1, m, kg);

  bias_relu_to_lds(Xl, acc0, acc1, b0, m, kg);

  // ---- layer 1 ----
  zero_acc2(acc0, acc1);
  gemm128x2(Xl, w1, acc0, acc1, m, kg);
  bias_relu_to_lds(Xl, acc0, acc1, b1, m, kg);

  // ---- layer 2 (no relu) + scatter-add onto receivers ----
  zero_acc2(acc0, acc1);
  gemm128x2(Xl, w2, acc0, acc1, m, kg);

#pragma unroll
  for (int st = 0; st < 2; ++st) {
    const v8f* acc = st ? acc1 : acc0;
#pragma unroll
    for (int v = 0; v < 8; ++v) {
      const int row = st * 16 + v + 8 * kg;
      const size_t rbase = (size_t)receivers[e0 + row] * D_FEAT;
#pragma unroll
      for (int nt = 0; nt < 8; ++nt) {
        const int n = nt * 16 + m;
        (void)__hip_atomic_fetch_add(&agg[rbase + n], acc[nt][v] + b2[n],
                                     __ATOMIC_RELAXED, __HIP_MEMORY_SCOPE_AGENT);
      }
    }
  }
}

// ---------------------------------------------------------------------------
__global__ void __launch_bounds__(BLOCK_THREADS)
node_update_kernel(const float* __restrict__ nodes, const float* __restrict__ agg,
                   const float* __restrict__ w0, const float* __restrict__ b0,
                   const float* __restrict__ w1, const float* __restrict__ b1,
                   const float* __restrict__ w2, const float* __restrict__ b2,
                   const float* __restrict__ wn, const float* __restrict__ ln_s,
                   const float* __restrict__ ln_b, float* __restrict__ out) {
  extern __shared__ float smem[];
  const int lane = threadIdx.x & 31;
  const int wave = threadIdx.x >> 5;
  float* Xl = smem + wave * (ROWS_PER_WAVE * TILE_STRIDE);

  const int tile = blockIdx.x * WAVES_PER_BLOCK + wave;
  if (tile * ROWS_PER_WAVE >= N_NODES) return;            // wave-uniform
  const int r0 = tile * ROWS_PER_WAVE;
  const int m = lane & 15;
  const int kg = lane >> 4;

  v8f acc0[8], acc1[8];
  zero_acc2(acc0, acc1);

  // ---- layer 0, part 0: node features (K = 0..127 of node_w0) ----
  for (int r = 0; r < ROWS_PER_WAVE; ++r) {
    int gr = r0 + r; gr = gr < N_NODES ? gr : N_NODES - 1;   // clamp ragged tail
    const v4f* src = (const v4f*)(nodes + (size_t)gr * D_FEAT);
    ((v4f*)(Xl + r * TILE_STRIDE))[lane] = src[lane];
  }
  gemm128x2(Xl, w0, acc0, acc1, m, kg);

  // ---- layer 0, part 1: aggregated messages (K = 128..255) ----
  for (int r = 0; r < ROWS_PER_WAVE; ++r) {
    int gr = r0 + r; gr = gr < N_NODES ? gr : N_NODES - 1;
    const v4f* src = (const v4f*)(agg + (size_t)gr * D_FEAT);
    ((v4f*)(Xl + r * TILE_STRIDE))[lane] = src[lane];
  }
  gemm128x2(Xl, w0 + 128 * 128, acc0, acc1, m, kg);

  bias_relu_to_lds(Xl, acc0, acc1, b0, m, kg);

  // ---- layer 1 ----
  zero_acc2(acc0, acc1);
  gemm128x2(Xl, w1, acc0, acc1, m, kg);
  bias_relu_to_lds(Xl, acc0, acc1, b1, m, kg);

  // ---- layer 2 + residual projection nodes @ w_node ----
  zero_acc2(acc0, acc1);
  gemm128x2(Xl, w2, acc0, acc1, m, kg);
  for (int r = 0; r < ROWS_PER_WAVE; ++r) {       // re-stage node tile for residual
    int gr = r0 + r; gr = gr < N_NODES ? gr : N_NODES - 1;
    const v4f* src = (const v4f*)(nodes + (size_t)gr * D_FEAT);
    ((v4f*)(Xl + r * TILE_STRIDE))[lane] = src[lane];
  }
  gemm128x2(Xl, wn, acc0, acc1, m, kg);

  // ---- pre-LN values (plus final bias b2) back to LDS ----
#pragma unroll
  for (int nt = 0; nt < 8; ++nt) {
    const int n = nt * 16 + m;
    const float bb = b2[n];
#pragma unroll
    for (int v = 0; v < 8; ++v) {
      Xl[(v + 8 * kg) * TILE_STRIDE + n]      = acc0[nt][v] + bb;
      Xl[(16 + v + 8 * kg) * TILE_STRIDE + n] = acc1[nt][v] + bb;
    }
  }

  // ---- LayerNorm: lane handles row (st*16 + m), half kg ----
#pragma unroll
  for (int st = 0; st < 2; ++st) {
    const int rl = st * 16 + m;
    const int gr = r0 + rl;
    float s = 0.f, sq = 0.f;
    const float* rowp = Xl + rl * TILE_STRIDE + kg * 64;
#pragma unroll
    for (int j4 = 0; j4 < 16; ++j4) {
      const v4f x = ((const v4f*)rowp)[j4];
      s  += x.x + x.y + x.z + x.w;
      sq += x.x * x.x + x.y * x.y + x.z * x.z + x.w * x.w;
    }
    s  += __shfl_xor(s, 16, 32);
    sq += __shfl_xor(sq, 16, 32);
    const float mean = s * (1.0f / 128.0f);
    const float var  = sq * (1.0f / 128.0f) - mean * mean;
    const float inv  = rsqrtf(var + 1e-6f);

    if (gr < N_NODES) {                     // ragged-tail guard (no WMMA below)
      v4f* op = (v4f*)(out + (size_t)gr * D_FEAT + kg * 64);
      const v4f* sp = (const v4f*)(ln_s + kg * 64);
      const v4f* bp = (const v4f*)(ln_b + kg * 64);
#pragma unroll
      for (int j4 = 0; j4 < 16; ++j4) {
        const v4f x = ((const v4f*)rowp)[j4];
        const v4f g = sp[j4];
        const v4f bb = bp[j4];
        v4f y;
        y.x = (x.x - mean) * inv * g.x + bb.x;
        y.y = (x.y - mean) * inv * g.y + bb.y;
        y.z = (x.z - mean) * inv * g.z + bb.z;
        y.w = (x.w - mean) * inv * g.w + bb.w;
        op[j4] = y;
      }
    }
  }
}

// ---------------------------------------------------------------------------
extern "C" void kernel_launch(void* const* d_in, const int* in_sizes, int n_in,
                              void* d_out, int out_size, void* d_ws, size_t ws_size,
                              hipStream_t stream) {
  const float* nodes     = (const float*)d_in[0];
  const float* edges     = (const float*)d_in[1];
  const int*   senders   = (const int*)d_in[2];
  const int*   receivers = (const int*)d_in[3];
  const float* msg_w0 = (const float*)d_in[4];
  const float* msg_b0 = (const float*)d_in[5];
  const float* msg_w1 = (const float*)d_in[6];
  const float* msg_b1 = (const float*)d_in[7];
  const float* msg_w2 = (const float*)d_in[8];
  const float* msg_b2 = (const float*)d_in[9];
  const float* node_w0 = (const float*)d_in[10];
  const float* node_b0 = (const float*)d_in[11];
  const float* node_w1 = (const float*)d_in[12];
  const float* node_b1 = (const float*)d_in[13];
  const float* node_w2 = (const float*)d_in[14];
  const float* node_b2 = (const float*)d_in[15];
  const float* w_node  = (const float*)d_in[16];
  const float* ln_s    = (const float*)d_in[17];
  const float* ln_b    = (const float*)d_in[18];

  float* agg = (float*)d_ws;                       // [N_NODES, D] f32 scratch
  float* out = (float*)d_out;

  // 1) zero the aggregation buffer (per-launch: deterministic).
  {
    const int n4 = N_NODES * D_FEAT / 4;           // 320000 float4
    zero_ws_kernel<<<(n4 + 255) / 256, 256, 0, stream>>>((v4f*)agg, n4);
  }

  // 2) edge MLP + scatter-add.  20000 tiles of 32 edges, 2 waves/block.
  {
    const int blocks = (N_EDGES / ROWS_PER_WAVE) / WAVES_PER_BLOCK;   // 10000
    edge_msg_kernel<<<blocks, BLOCK_THREADS, SMEM_BYTES, stream>>>(
        nodes, edges, senders, receivers,
        msg_w0, msg_b0, msg_w1, msg_b1, msg_w2, msg_b2, agg);
  }

  // 3) node MLP + residual + LayerNorm.  313 tiles of 32 nodes (clamped tail).
  {
    const int tiles = (N_NODES + ROWS_PER_WAVE - 1) / ROWS_PER_WAVE;  // 313
    const int blocks = (tiles + WAVES_PER_BLOCK - 1) / WAVES_PER_BLOCK; // 157
    node_update_kernel<<<blocks, BLOCK_THREADS, SMEM_BYTES, stream>>>(
        nodes, agg,
        node_w0, node_b0, node_w1, node_b1, node_w2, node_b2,
        w_node, ln_s, ln_b, out);
  }
}